// VGCNEncoder_62036507623793
// MI455X (gfx1250) — compile-verified
//
#include <hip/hip_runtime.h>
#include <hip/hip_bf16.h>

typedef __attribute__((ext_vector_type(2))) float v2f;
typedef __attribute__((ext_vector_type(8))) float v8f;

#define IN_F  256
#define HID_F 256
#define OUT_F 128

// ---------------- degree / normalization ----------------
__global__ void k_deg_init(float* __restrict__ deg, int n) {
    int i = blockIdx.x * blockDim.x + threadIdx.x;
    if (i < n) deg[i] = 1.0f;                 // self-loop contribution
}

__global__ void k_deg_count(const int* __restrict__ dst, float* __restrict__ deg, int e) {
    int i = blockIdx.x * blockDim.x + threadIdx.x;
    if (i < e) atomicAdd(&deg[dst[i]], 1.0f);
}

__global__ void k_dinv(float* __restrict__ deg, int n) {
    int i = blockIdx.x * blockDim.x + threadIdx.x;
    if (i < n) deg[i] = rsqrtf(deg[i]);       // deg >= 1 always (self-loop)
}

// ---------------- fp32 WMMA GEMM: C[M,256] = f(A[M,256]) @ B[256, 256] ----------------
// Each wave computes a 16x64 strip of C via V_WMMA_F32_16X16X4_F32.
// RELU_A: apply relu to A elements at load (bias was folded into A at agg-init).
// SPLIT_B: columns 0..127 come from B0[256,128], columns 128..255 from B1[256,128].
template <bool RELU_A, bool SPLIT_B>
__global__ __launch_bounds__(256)
void k_gemm(const float* __restrict__ A, const float* __restrict__ B0,
            const float* __restrict__ B1, float* __restrict__ C, int M) {
    const int lane = threadIdx.x & 31;
    const int wid  = (int)((blockIdx.x * blockDim.x + threadIdx.x) >> 5);
    const int mtiles = (M + 15) >> 4;
    if (wid >= mtiles * 4) return;            // uniform per wave -> EXEC all-1s below

    const int mbase = (wid >> 2) * 16;
    const int nbase = (wid & 3) * 64;
    const int r16   = lane & 15;
    const int khalf = (lane >> 4) << 1;       // 0 for lanes 0-15, 2 for lanes 16-31

    int arowi = mbase + r16;                  // clamp for ragged M (never taken here)
    if (arowi > M - 1) arowi = M - 1;
    const float* __restrict__ arow = A + (size_t)arowi * HID_F;

    v8f acc[4] = {};

    for (int k = 0; k < HID_F; k += 4) {
        // A fragment: 16x4 f32 (lane half selects K pair)
        v2f a;
        a.x = arow[k + khalf + 0];
        a.y = arow[k + khalf + 1];
        if (RELU_A) { a.x = fmaxf(a.x, 0.0f); a.y = fmaxf(a.y, 0.0f); }

#pragma unroll
        for (int j = 0; j < 4; ++j) {
            const int g = nbase + j * 16 + r16;   // global output column
            v2f b;
            if (SPLIT_B) {
                const float* __restrict__ Bp = (g < OUT_F) ? B0 : B1;
                const int cc = (g < OUT_F) ? g : (g - OUT_F);
                b.x = Bp[(size_t)(k + khalf + 0) * OUT_F + cc];
                b.y = Bp[(size_t)(k + khalf + 1) * OUT_F + cc];
            } else {
                b.x = B0[(size_t)(k + khalf + 0) * HID_F + g];
                b.y = B0[(size_t)(k + khalf + 1) * HID_F + g];
            }
            acc[j] = __builtin_amdgcn_wmma_f32_16x16x4_f32(
                false, a, false, b, (short)0, acc[j], false, false);
        }
    }

    // C/D layout: VGPR r -> row (mbase + r + 8*(lane>=16)), col = nbase + j*16 + (lane&15).
    // Tile validity is wave-uniform: hoist the bounds check so the common path is a
    // branch-free clause of 32 global_store_b32 with immediate row offsets.
    const int rowhalf = (lane >> 4) << 3;     // 0 or 8
    if (mbase + 16 <= M) {                    // full tile (always true for M%16==0)
#pragma unroll
        for (int j = 0; j < 4; ++j) {
            float* __restrict__ cp =
                C + (size_t)(mbase + rowhalf) * HID_F + (nbase + j * 16 + r16);
#pragma unroll
            for (int r = 0; r < 8; ++r)
                cp[(size_t)r * HID_F] = acc[j][r];
        }
    } else {                                  // ragged tail (generic fallback)
#pragma unroll
        for (int j = 0; j < 4; ++j) {
            const int col = nbase + j * 16 + r16;
#pragma unroll
            for (int r = 0; r < 8; ++r) {
                const int row = mbase + rowhalf + r;
                if (row < M) C[(size_t)row * HID_F + col] = acc[j][r];
            }
        }
    }
}

// ---------------- aggregation pass 1 (hidden layer) ----------------
// init: agg[i][f] = b1[f] + Y1[i][f] * dinv[i]^2   (bias + self-loop, non-atomic)
__global__ void k_init_agg1(const float* __restrict__ Y1, const float* __restrict__ dinv,
                            const float* __restrict__ b1, float* __restrict__ agg, int n) {
    int t = blockIdx.x * blockDim.x + threadIdx.x;
    if (t >= n * HID_F) return;
    int i = t >> 8, f = t & 255;
    float di = dinv[i];
    agg[t] = b1[f] + Y1[t] * di * di;
}

// one wave32 per edge: agg[dst] += Y1[src] * (dinv[src]*dinv[dst])
__global__ __launch_bounds__(256)
void k_edge_agg1(const float* __restrict__ Y, const int* __restrict__ src,
                 const int* __restrict__ dst, const float* __restrict__ dinv,
                 float* __restrict__ agg, int e) {
    const int lane = threadIdx.x & 31;
    const int w = (int)((blockIdx.x * blockDim.x + threadIdx.x) >> 5);
    if (w >= e) return;
    const int s = src[w], d = dst[w];
    const float nrm = dinv[s] * dinv[d];
    const float* __restrict__ yr = Y + (size_t)s * HID_F;
    float* __restrict__ ar = agg + (size_t)d * HID_F;
#pragma unroll
    for (int f = lane; f < HID_F; f += 32)
        atomicAdd(&ar[f], yr[f] * nrm);
}

// ---------------- aggregation pass 2 (directly into d_out, split mu/ls) ----------------
__global__ void k_init_out(const float* __restrict__ Y2, const float* __restrict__ dinv,
                           const float* __restrict__ bmu, const float* __restrict__ bls,
                           float* __restrict__ out, int n) {
    int t = blockIdx.x * blockDim.x + threadIdx.x;
    if (t >= n * HID_F) return;
    int i = t >> 8, f = t & 255;
    float di = dinv[i];
    float v = Y2[t] * di * di;
    if (f < OUT_F) out[(size_t)i * OUT_F + f] = v + bmu[f];
    else out[(size_t)n * OUT_F + (size_t)i * OUT_F + (f - OUT_F)] = v + bls[f - OUT_F];
}

__global__ __launch_bounds__(256)
void k_edge_agg2(const float* __restrict__ Y2, const int* __restrict__ src,
                 const int* __restrict__ dst, const float* __restrict__ dinv,
                 float* __restrict__ out, int e, int n) {
    const int lane = threadIdx.x & 31;
    const int w = (int)((blockIdx.x * blockDim.x + threadIdx.x) >> 5);
    if (w >= e) return;
    const int s = src[w], d = dst[w];
    const float nrm = dinv[s] * dinv[d];
    const float* __restrict__ yr = Y2 + (size_t)s * HID_F;
    float* __restrict__ mu = out + (size_t)d * OUT_F;
    float* __restrict__ ls = out + (size_t)n * OUT_F + (size_t)d * OUT_F;
#pragma unroll
    for (int f = lane; f < HID_F; f += 32) {
        const float v = yr[f] * nrm;
        if (f < OUT_F) atomicAdd(&mu[f], v);
        else           atomicAdd(&ls[f - OUT_F], v);
    }
}

// ---------------- launcher ----------------
extern "C" void kernel_launch(void* const* d_in, const int* in_sizes, int n_in,
                              void* d_out, int out_size, void* d_ws, size_t ws_size,
                              hipStream_t stream) {
    const float* x   = (const float*)d_in[0];
    const int*   ei  = (const int*)  d_in[1];
    const float* W1  = (const float*)d_in[2];
    const float* b1  = (const float*)d_in[3];
    const float* Wmu = (const float*)d_in[4];
    const float* bmu = (const float*)d_in[5];
    const float* Wls = (const float*)d_in[6];
    const float* bls = (const float*)d_in[7];
    float* out = (float*)d_out;

    const int N = in_sizes[0] / IN_F;   // 50000
    const int E = in_sizes[1] / 2;      // 800000
    const int* src = ei;                // edge_index[0]
    const int* dst = ei + E;            // edge_index[1]

    // workspace carve: dinv[N] | bufA[N*256] (Y1 then Y2) | bufB[N*256] (agg1)
    float* dinv = (float*)d_ws;
    size_t off = (((size_t)N * sizeof(float)) + 255) & ~(size_t)255;
    float* bufA = (float*)((char*)d_ws + off);
    float* bufB = bufA + (size_t)N * HID_F;

    const int B = 256;
    // 1) degrees -> dinv
    k_deg_init <<<(N + B - 1) / B, B, 0, stream>>>(dinv, N);
    k_deg_count<<<(E + B - 1) / B, B, 0, stream>>>(dst, dinv, E);
    k_dinv     <<<(N + B - 1) / B, B, 0, stream>>>(dinv, N);

    const int mtiles = (N + 15) / 16;
    const int gemm_threads = mtiles * 4 * 32;
    const int gemm_blocks  = (gemm_threads + B - 1) / B;

    // 2) Y1 = X @ W1   (fp32 WMMA)
    k_gemm<false, false><<<gemm_blocks, B, 0, stream>>>(x, W1, nullptr, bufA, N);

    // 3) agg1 = b1 + self + scatter(Y1[src]*norm)
    k_init_agg1<<<(N * HID_F + B - 1) / B, B, 0, stream>>>(bufA, dinv, b1, bufB, N);
    k_edge_agg1<<<((size_t)E * 32 + B - 1) / B, B, 0, stream>>>(bufA, src, dst, dinv, bufB, E);

    // 4) Y2 = relu(agg1) @ [Wmu | Wls]   (relu fused into A-load; overwrite bufA)
    k_gemm<true, true><<<gemm_blocks, B, 0, stream>>>(bufB, Wmu, Wls, bufA, N);

    // 5) out = bias + self + scatter(Y2[src]*norm), split mu/ls layout
    k_init_out <<<(N * HID_F + B - 1) / B, B, 0, stream>>>(bufA, dinv, bmu, bls, out, N);
    k_edge_agg2<<<((size_t)E * 32 + B - 1) / B, B, 0, stream>>>(bufA, src, dst, dinv, out, E, N);
}